// LowRankRayTracer_35742717837730
// MI455X (gfx1250) — compile-verified
//
#include <hip/hip_runtime.h>

typedef float v2f __attribute__((ext_vector_type(2)));
typedef float v8f __attribute__((ext_vector_type(8)));

#define D_DIRS 1296
#define K_PTS  64
#define RANK   16
#define F_SUB  408
#define N_FTILES 26           // ceil(408/16)
#define DT     (200.0f / 64.0f)

__global__ void csi_zero_kernel(float* __restrict__ out) {
    int i = blockIdx.x * blockDim.x + threadIdx.x;
    if (i < 2 * F_SUB) out[i] = 0.0f;
}

__global__ __launch_bounds__(256) void csi_wmma_kernel(
    const float* __restrict__ attR, const float* __restrict__ attI,
    const float* __restrict__ radR, const float* __restrict__ radI,
    const float* __restrict__ frR,  const float* __restrict__ frI,
    float* __restrict__ out)
{
    // A-side staging: [0]=radR [1]=radI [2]=attR [3]=attI [4]=csR*dt [5]=csI*dt
    __shared__ float sA[6][K_PTS * RANK];       // 24.6 KB

    const int d   = blockIdx.x;
    const int tid = threadIdx.x;
    const int S   = K_PTS * RANK;               // 1024 contiguous floats per dir

    const float* g0 = radR + (size_t)d * S;
    const float* g1 = radI + (size_t)d * S;
    const float* g2 = attR + (size_t)d * S;
    const float* g3 = attI + (size_t)d * S;
    for (int i = tid; i < S; i += 256) {
        sA[0][i] = g0[i];
        sA[1][i] = g1[i];
        sA[2][i] = g2[i];
        sA[3][i] = g3[i];
    }
    __syncthreads();

    // conj(u_hat_rho) = cumsum_k(att) * dt  -> sA[4]/sA[5]
    if (tid < 32) {
        const int r   = tid & 15;
        const int sel = tid >> 4;               // 0 = re, 1 = im
        float run = 0.0f;
        for (int k = 0; k < K_PTS; ++k) {
            run += sA[2 + sel][k * RANK + r];
            sA[4 + sel][k * RANK + r] = run * DT;
        }
    }
    __syncthreads();

    const int wave = tid >> 5;
    const int lane = tid & 31;
    const int ln   = lane & 15;                 // N column / A row within tile
    const int hi   = lane >> 4;                 // half-wave select
    const float scale = DT / (float)D_DIRS;

    #pragma unroll 1
    for (int ft = wave; ft < N_FTILES; ft += 8) {
        const int  fcol   = ft * 16 + ln;
        const bool fvalid = (fcol < F_SUB);
        const int  fc     = fvalid ? fcol : (F_SUB - 1);

        // B fragments depend only on (ft, ks): preload all 4 rank-steps.
        // B[r][f] = freq[f][r], freq is (F,16) row-major.
        v2f bR[4], bI[4];
        #pragma unroll
        for (int ks = 0; ks < 4; ++ks) {
            const int Kc = ks * 4 + hi * 2;
            bR[ks].x = frR[fc * RANK + Kc];
            bR[ks].y = frR[fc * RANK + Kc + 1];
            bI[ks].x = frI[fc * RANK + Kc];
            bI[ks].y = frI[fc * RANK + Kc + 1];
            if (!fvalid) {
                bR[ks].x = 0.f; bR[ks].y = 0.f;
                bI[ks].x = 0.f; bI[ks].y = 0.f;
            }
        }

        float accRe = 0.0f, accIm = 0.0f;

        #pragma unroll 1
        for (int kt = 0; kt < 4; ++kt) {
            v8f Pr = {}, Pi = {}, Qr = {}, Qi = {}, Hr = {}, Hi = {};
            const int M = kt * 16 + ln;         // k-point row for A fragment

            #pragma unroll
            for (int ks = 0; ks < 4; ++ks) {
                const int Kc = ks * 4 + hi * 2; // rank slice for this lane half

                // A fragments from LDS: A[M][Kc], A[M][Kc+1]
                const int o = M * RANK + Kc;
                v2f aRr, aRi, aAr, aAi, aCr, aCi;
                aRr.x = sA[0][o]; aRr.y = sA[0][o + 1];
                aRi.x = sA[1][o]; aRi.y = sA[1][o + 1];
                aAr.x = sA[2][o]; aAr.y = sA[2][o + 1];
                aAi.x = sA[3][o]; aAi.y = sA[3][o + 1];
                aCr.x = sA[4][o]; aCr.y = sA[4][o + 1];
                aCi.x = sA[5][o]; aCi.y = sA[5][o + 1];
                v2f aRin; aRin.x = -aRi.x; aRin.y = -aRi.y;
                v2f aAin; aAin.x = -aAi.x; aAin.y = -aAi.y;
                v2f aCin; aCin.x = -aCi.x; aCin.y = -aCi.y;

                // p = conj(rad) . f  : re = rR.fR + rI.fI ; im = rR.fI - rI.fR
                Pr = __builtin_amdgcn_wmma_f32_16x16x4_f32(false, aRr,  false, bR[ks], (short)0, Pr, false, false);
                Pr = __builtin_amdgcn_wmma_f32_16x16x4_f32(false, aRi,  false, bI[ks], (short)0, Pr, false, false);
                Pi = __builtin_amdgcn_wmma_f32_16x16x4_f32(false, aRr,  false, bI[ks], (short)0, Pi, false, false);
                Pi = __builtin_amdgcn_wmma_f32_16x16x4_f32(false, aRin, false, bR[ks], (short)0, Pi, false, false);
                // q = conj(att) . f
                Qr = __builtin_amdgcn_wmma_f32_16x16x4_f32(false, aAr,  false, bR[ks], (short)0, Qr, false, false);
                Qr = __builtin_amdgcn_wmma_f32_16x16x4_f32(false, aAi,  false, bI[ks], (short)0, Qr, false, false);
                Qi = __builtin_amdgcn_wmma_f32_16x16x4_f32(false, aAr,  false, bI[ks], (short)0, Qi, false, false);
                Qi = __builtin_amdgcn_wmma_f32_16x16x4_f32(false, aAin, false, bR[ks], (short)0, Qi, false, false);
                // h = (cumsum(att)*dt) . f : re = cR.fR - cI.fI ; im = cR.fI + cI.fR
                Hr = __builtin_amdgcn_wmma_f32_16x16x4_f32(false, aCr,  false, bR[ks], (short)0, Hr, false, false);
                Hr = __builtin_amdgcn_wmma_f32_16x16x4_f32(false, aCin, false, bI[ks], (short)0, Hr, false, false);
                Hi = __builtin_amdgcn_wmma_f32_16x16x4_f32(false, aCr,  false, bI[ks], (short)0, Hi, false, false);
                Hi = __builtin_amdgcn_wmma_f32_16x16x4_f32(false, aCi,  false, bR[ks], (short)0, Hi, false, false);
            }

            // C layout: element i of this lane = C[M = i + 8*hi][N = ln]
            #pragma unroll
            for (int i = 0; i < 8; ++i) {
                const int k = kt * 16 + i + hi * 8;
                const float pr = Pr[i], pim = Pi[i];
                const float qr = Qr[i], qim = Qi[i];
                const float pqr = pr * qr - pim * qim;
                const float pqi = pr * qim + pim * qr;
                const float hr  = (k >= 1) ? Hr[i] : 0.0f;
                const float him = (k >= 1) ? Hi[i] : 0.0f;
                accRe += pqr * (1.0f + hr) - pqi * him;
                accIm += pqr * him + pqi * (1.0f + hr);
            }
        }

        accRe *= scale;
        accIm *= scale;
        // fold the two half-wave M-groups (lanes L and L+16 share the same f)
        accRe += __shfl_xor(accRe, 16, 32);
        accIm += __shfl_xor(accIm, 16, 32);

        if (hi == 0 && fvalid) {
            atomicAdd(&out[2 * fcol + 0], accRe);
            atomicAdd(&out[2 * fcol + 1], accIm);
        }
    }
}

extern "C" void kernel_launch(void* const* d_in, const int* in_sizes, int n_in,
                              void* d_out, int out_size, void* d_ws, size_t ws_size,
                              hipStream_t stream) {
    (void)in_sizes; (void)n_in; (void)out_size; (void)d_ws; (void)ws_size;
    const float* attR = (const float*)d_in[0];
    const float* attI = (const float*)d_in[1];
    const float* radR = (const float*)d_in[2];
    const float* radI = (const float*)d_in[3];
    const float* frR  = (const float*)d_in[4];
    const float* frI  = (const float*)d_in[5];
    float* out = (float*)d_out;

    csi_zero_kernel<<<(2 * F_SUB + 255) / 256, 256, 0, stream>>>(out);
    csi_wmma_kernel<<<D_DIRS, 256, 0, stream>>>(attR, attI, radR, radI, frR, frI, out);
}